// SparseShiftGateRecurrence_88716844466862
// MI455X (gfx1250) — compile-verified
//
#include <hip/hip_runtime.h>

// Problem constants (from reference): x[8, 4096, 2048] f32, active = 512 channels.
#define B_      8
#define S_      4096
#define D_      2048
#define A_      512
#define PASS_   (D_ - A_)          // 1536 passthrough channels
#define CHUNK   32                 // timesteps per chunk
#define NCHUNK  (S_ / CHUNK)       // 128 chunks
#define TPB     256                // 8 wave32 per block

// Native clang vectors (HIP float2/float4 are structs; the nontemporal
// builtins only accept scalar/native-vector pointees).
typedef float f2 __attribute__((ext_vector_type(2)));
typedef float f4 __attribute__((ext_vector_type(4)));

// ---------------------------------------------------------------------------
// CDNA5 async global->LDS copy (ASYNCcnt path).
// Round-1 probe confirmed signature: (v4i addrspace(1)*, v4i addrspace(3)*, imm, imm).
// ---------------------------------------------------------------------------
typedef int v4i __attribute__((ext_vector_type(4)));
typedef __attribute__((address_space(1))) v4i* gp128_t;   // global int4*
typedef __attribute__((address_space(3))) v4i* lp128_t;   // LDS int4*

__device__ __forceinline__ void async_copy_b128(const void* gsrc, void* lds_dst) {
#if defined(__HIP_DEVICE_COMPILE__)
#if __has_builtin(__builtin_amdgcn_global_load_async_to_lds_b128)
  __builtin_amdgcn_global_load_async_to_lds_b128(
      (gp128_t)(unsigned long long)gsrc, (lp128_t)lds_dst,
      /*offset=*/0, /*cpol=*/0);
#else
  unsigned loff = (unsigned)(unsigned long long)lds_dst;
  asm volatile("global_load_async_to_lds_b128 %0, %1, off"
               :: "v"(loff), "v"(gsrc) : "memory");
#endif
#endif
}

__device__ __forceinline__ void async_wait_all() {
#if defined(__HIP_DEVICE_COMPILE__)
#if __has_builtin(__builtin_amdgcn_s_wait_asynccnt)
  __builtin_amdgcn_s_wait_asynccnt(0);
#else
  asm volatile("s_wait_asynccnt 0x0" ::: "memory");
#endif
#endif
}

// Stage x[b, c*CHUNK:(c+1)*CHUNK, 0:A_] (32 x 512 f32 = 64 KB) into LDS tile.
// Regular temporal hint: active x (64 MB total) should persist in the 192 MB L2
// so the phase-3 re-read hits L2 instead of HBM.
__device__ __forceinline__ void stage_tile_async(const float* gbase, float* tile,
                                                 int tid) {
  // 4096 16-byte units; 128 units per 512-float row.
#pragma unroll 4
  for (int u = tid; u < (CHUNK * A_) / 4; u += TPB) {
    const int row = u >> 7;
    const int col = (u & 127) << 2;               // float index within row
    async_copy_b128(gbase + (size_t)row * D_ + col, tile + (size_t)u * 4);
  }
}

// ---------------------------------------------------------------------------
// Phase 1: per (b, chunk) compute chunk-local final state with h0 = 0.
//   carry[b, c, a] = sum_{i<CHUNK} alpha^(CHUNK-1-i) * beta * x[b, c*CHUNK+i, a]
// ---------------------------------------------------------------------------
__global__ __launch_bounds__(TPB) void sg_chunk_carry(
    const float* __restrict__ x, const float* __restrict__ alpha,
    const float* __restrict__ beta, float* __restrict__ carry) {
  __shared__ float tile[CHUNK * A_];
  const int bc  = blockIdx.x;            // 0 .. B_*NCHUNK-1
  const int b   = bc / NCHUNK;
  const int c   = bc % NCHUNK;
  const int tid = threadIdx.x;

  const float* gbase = x + (size_t)b * S_ * D_ + (size_t)c * CHUNK * D_;
  stage_tile_async(gbase, tile, tid);

  const int a0 = tid * 2;                          // 2 channels per thread
  const f2 al = *(const f2*)(alpha + a0);
  const f2 be = *(const f2*)(beta + a0);

  async_wait_all();
  __syncthreads();

  float h0 = 0.0f, h1 = 0.0f;
#pragma unroll
  for (int t = 0; t < CHUNK; ++t) {
    const f2 xv = *(const f2*)&tile[t * A_ + a0];  // bank-conflict-free
    h0 = fmaf(al.x, h0, be.x * xv.x);
    h1 = fmaf(al.y, h1, be.y * xv.y);
  }
  f2 s; s.x = h0; s.y = h1;
  *(f2*)(carry + ((size_t)b * NCHUNK + c) * A_ + a0) = s;
}

// ---------------------------------------------------------------------------
// Phase 2: per (b, a) serial combine over 128 chunks.
//   prefix[b, c, a] = state entering chunk c;  h <- alpha^CHUNK * h + carry
// ---------------------------------------------------------------------------
__global__ __launch_bounds__(TPB) void sg_chunk_prefix(
    const float* __restrict__ alpha, const float* __restrict__ carry,
    float* __restrict__ prefix) {
  const int idx = blockIdx.x * TPB + threadIdx.x;  // 0 .. B_*A_-1
  const int b = idx / A_;
  const int a = idx % A_;

  const float al = alpha[a];
  const float a2  = al * al;
  const float a4  = a2 * a2;
  const float a8  = a4 * a4;
  const float a16 = a8 * a8;
  const float aL  = a16 * a16;                     // alpha^32 == alpha^CHUNK

  float h = 0.0f;
  const size_t base = (size_t)b * NCHUNK * A_ + a;
  for (int c = 0; c < NCHUNK; ++c) {
    const size_t off = base + (size_t)c * A_;
    prefix[off] = h;
    h = fmaf(aL, h, carry[off]);
  }
}

// ---------------------------------------------------------------------------
// Phase 3: per (b, chunk) replay scan seeded by prefix, write active outputs;
// passthrough channels copied (non-temporal both ways) while the async tile
// load is in flight. All d_out stores are non-temporal: output is never
// re-read, so don't let 256 MB of stores evict the active-x L2 working set.
// ---------------------------------------------------------------------------
__global__ __launch_bounds__(TPB) void sg_chunk_scan(
    const float* __restrict__ x, const float* __restrict__ alpha,
    const float* __restrict__ beta, const float* __restrict__ prefix,
    float* __restrict__ out) {
  __shared__ float tile[CHUNK * A_];
  const int bc  = blockIdx.x;
  const int b   = bc / NCHUNK;
  const int c   = bc % NCHUNK;
  const int tid = threadIdx.x;

  const size_t row_base = (size_t)b * S_ * D_ + (size_t)c * CHUNK * D_;
  stage_tile_async(x + row_base, tile, tid);

  // Passthrough copy overlaps with the async tile transfer (streaming, NT).
  {
    const float* pin  = x   + row_base + A_;
    float*       pout = out + row_base + A_;
    const int RU = PASS_ / 4;                      // 384 f4 per row
#pragma unroll 2
    for (int u = tid; u < CHUNK * RU; u += TPB) {
      const int row = u / RU;
      const int col = (u % RU) * 4;
      const f4 v =
          __builtin_nontemporal_load((const f4*)(pin + (size_t)row * D_ + col));
      __builtin_nontemporal_store(v, (f4*)(pout + (size_t)row * D_ + col));
    }
  }

  const int a0 = tid * 2;
  const f2 al = *(const f2*)(alpha + a0);
  const f2 be = *(const f2*)(beta + a0);
  f2 h = *(const f2*)(prefix + ((size_t)b * NCHUNK + c) * A_ + a0);

  async_wait_all();
  __syncthreads();

  float* obase = out + row_base + a0;
#pragma unroll
  for (int t = 0; t < CHUNK; ++t) {
    const f2 xv = *(const f2*)&tile[t * A_ + a0];
    h.x = fmaf(al.x, h.x, be.x * xv.x);
    h.y = fmaf(al.y, h.y, be.y * xv.y);
    __builtin_nontemporal_store(h, (f2*)(obase + (size_t)t * D_));  // 256B/wave
  }
}

// ---------------------------------------------------------------------------
extern "C" void kernel_launch(void* const* d_in, const int* in_sizes, int n_in,
                              void* d_out, int out_size, void* d_ws, size_t ws_size,
                              hipStream_t stream) {
  const float* x     = (const float*)d_in[0];
  const float* alpha = (const float*)d_in[1];
  const float* beta  = (const float*)d_in[2];
  float*       out   = (float*)d_out;

  // Workspace: carry (2 MB) + prefix (2 MB), both fully written before read.
  float* carry  = (float*)d_ws;
  float* prefix = carry + (size_t)B_ * NCHUNK * A_;

  sg_chunk_carry <<<B_ * NCHUNK,        TPB, 0, stream>>>(x, alpha, beta, carry);
  sg_chunk_prefix<<<(B_ * A_) / TPB,    TPB, 0, stream>>>(alpha, carry, prefix);
  sg_chunk_scan  <<<B_ * NCHUNK,        TPB, 0, stream>>>(x, alpha, beta, prefix, out);
}